// FRPredictHeadWithFlatten_49460843381559
// MI455X (gfx1250) — compile-verified
//
#include <hip/hip_runtime.h>

typedef __attribute__((ext_vector_type(2))) float v2f;
typedef __attribute__((ext_vector_type(8))) float v8f;

#define WAY   20
#define SHOT  5
#define CDIM  1024
#define NCLS  21            // way + 1
#define NBOX  84            // NUM_CLASSES * 4
#define ROI   2048
#define SBROWS 105          // NCLS * SHOT
#define SBPAD  112          // padded to multiple of 16

// ---------------------------------------------------------------------------
// Generic f32 WMMA GEMM:  OUT[M,N] = X[M,K] @ W[N,K]^T (+ bias)
// K is a compile-time constant (1024 everywhere here), multiple of 16.
// Block = 256 threads = 8 waves; wave tile = 32x64 (2x4 fragments of 16x16);
// block tile = 128x128. grid = (ceil(N/128), ceil(M/128)).
// v_wmma_f32_16x16x4_f32 fragment layout (wave32):
//   A (16x4): lane<16 -> row M=lane,    a.x=K(k0),   a.y=K(k0+1)
//             lane>=16-> row M=lane-16, a.x=K(k0+2), a.y=K(k0+3)
//   B (4x16): lane<16 -> col N=lane,    b.x=B[k0][n],   b.y=B[k0+1][n]
//             lane>=16-> col N=lane-16, b.x=B[k0+2][n], b.y=B[k0+3][n]
//   (B element (k,n) == W[n][k]: contiguous 8B loads along K for both.)
//   C/D (16x16): lane<16 -> col=lane, acc[i] = row i; lane>=16 -> row i+8
// The +2*hi K-offset is folded into the base pointers so the k-loop has a
// uniform trip count (keeps EXEC all-ones, enables unrolling + IOFFSET folding).
// ---------------------------------------------------------------------------
template <int K>
__global__ __launch_bounds__(256)
void gemm_xwt_kernel(const float* __restrict__ X, const float* __restrict__ W,
                     const float* __restrict__ bias, float* __restrict__ OUT,
                     int M, int N, int ldx, int ldw, int ldo)
{
    const int tid  = threadIdx.x;
    const int lane = tid & 31;
    const int wave = tid >> 5;
    const int lo   = lane & 15;
    const int hi   = lane >> 4;      // 0 or 1
    const int wm   = wave >> 1;      // 0..3  (M direction)
    const int wn   = wave & 1;       // 0..1  (N direction)

    const int m0 = blockIdx.y * 128 + wm * 32;
    const int n0 = blockIdx.x * 128 + wn * 64;

    // Row-clamped base pointers (stores masked; loads stay in bounds with
    // uniform control flow so EXEC is all-ones for every WMMA).
    const float* xa0 = X + (size_t)min(m0 + lo,      M - 1) * ldx + 2 * hi;
    const float* xa1 = X + (size_t)min(m0 + 16 + lo, M - 1) * ldx + 2 * hi;
    const float* wb0 = W + (size_t)min(n0 + lo,      N - 1) * ldw + 2 * hi;
    const float* wb1 = W + (size_t)min(n0 + 16 + lo, N - 1) * ldw + 2 * hi;
    const float* wb2 = W + (size_t)min(n0 + 32 + lo, N - 1) * ldw + 2 * hi;
    const float* wb3 = W + (size_t)min(n0 + 48 + lo, N - 1) * ldw + 2 * hi;

    v8f acc[2][4];
    #pragma unroll
    for (int mt = 0; mt < 2; ++mt)
        #pragma unroll
        for (int j = 0; j < 4; ++j)
            acc[mt][j] = (v8f)0.0f;

    // Uniform loop: K/16 iterations, 4 k-steps (of 4) per iteration.
    // Constant element offsets 0/4/8/12 fold into the load IOFFSET field.
    #pragma unroll 1
    for (int it = 0; it < K / 16; ++it) {
        #pragma unroll
        for (int u = 0; u < 4; ++u) {
            const int o = 4 * u;
            v2f a0 = *(const v2f*)(xa0 + o);
            v2f a1 = *(const v2f*)(xa1 + o);
            v2f b0 = *(const v2f*)(wb0 + o);
            v2f b1 = *(const v2f*)(wb1 + o);
            v2f b2 = *(const v2f*)(wb2 + o);
            v2f b3 = *(const v2f*)(wb3 + o);
            acc[0][0] = __builtin_amdgcn_wmma_f32_16x16x4_f32(false, a0, false, b0, (short)0, acc[0][0], false, false);
            acc[0][1] = __builtin_amdgcn_wmma_f32_16x16x4_f32(false, a0, false, b1, (short)0, acc[0][1], false, false);
            acc[0][2] = __builtin_amdgcn_wmma_f32_16x16x4_f32(false, a0, false, b2, (short)0, acc[0][2], false, false);
            acc[0][3] = __builtin_amdgcn_wmma_f32_16x16x4_f32(false, a0, false, b3, (short)0, acc[0][3], false, false);
            acc[1][0] = __builtin_amdgcn_wmma_f32_16x16x4_f32(false, a1, false, b0, (short)0, acc[1][0], false, false);
            acc[1][1] = __builtin_amdgcn_wmma_f32_16x16x4_f32(false, a1, false, b1, (short)0, acc[1][1], false, false);
            acc[1][2] = __builtin_amdgcn_wmma_f32_16x16x4_f32(false, a1, false, b2, (short)0, acc[1][2], false, false);
            acc[1][3] = __builtin_amdgcn_wmma_f32_16x16x4_f32(false, a1, false, b3, (short)0, acc[1][3], false, false);
        }
        xa0 += 16; xa1 += 16;
        wb0 += 16; wb1 += 16; wb2 += 16; wb3 += 16;
    }

    // Bias values for the 4 N-fragments (predicated; only 4 loads).
    float bj[4] = {0.0f, 0.0f, 0.0f, 0.0f};
    if (bias != nullptr) {
        #pragma unroll
        for (int j = 0; j < 4; ++j) {
            const int n = n0 + j * 16 + lo;
            if (n < N) bj[j] = bias[n];
        }
    }

    const int mbase = m0 + hi * 8;
    if (m0 + 32 <= M && n0 + 64 <= N) {
        // Fast path: whole wave tile in bounds -> unpredicated stores,
        // one row base per 4 stores (j*16 elems -> 64B immediate offsets).
        #pragma unroll
        for (int mt = 0; mt < 2; ++mt) {
            #pragma unroll
            for (int i = 0; i < 8; ++i) {
                float* o = OUT + (size_t)(mbase + mt * 16 + i) * ldo + n0 + lo;
                #pragma unroll
                for (int j = 0; j < 4; ++j)
                    o[j * 16] = acc[mt][j][i] + bj[j];
            }
        }
    } else {
        #pragma unroll
        for (int mt = 0; mt < 2; ++mt) {
            #pragma unroll
            for (int i = 0; i < 8; ++i) {
                const int m = mbase + mt * 16 + i;
                if (m >= M) continue;
                float* o = OUT + (size_t)m * ldo;
                #pragma unroll
                for (int j = 0; j < 4; ++j) {
                    const int n = n0 + j * 16 + lo;
                    if (n < N) o[n] = acc[mt][j][i] + bj[j];
                }
            }
        }
    }
}

// ---------------------------------------------------------------------------
// sb = concat(bg_mean (5xC), s (100xC)); rows [105,112) zeroed.
// bg_mean[t,c] = mean_w bg[(w*SHOT+t), c]
// ---------------------------------------------------------------------------
__global__ __launch_bounds__(256)
void build_sb_kernel(const float* __restrict__ s, const float* __restrict__ bg,
                     float* __restrict__ sb)
{
    const int idx = blockIdx.x * blockDim.x + threadIdx.x;
    if (idx >= SBPAD * CDIM) return;
    const int row = idx / CDIM;
    const int c   = idx % CDIM;
    float v = 0.0f;
    if (row < SHOT) {
        float acc = 0.0f;
        #pragma unroll
        for (int w = 0; w < WAY; ++w)
            acc += bg[(size_t)(w * SHOT + row) * CDIM + c];
        v = acc * (1.0f / (float)WAY);
    } else if (row < SHOT + WAY * SHOT) {
        v = s[(size_t)(row - SHOT) * CDIM + c];
    }
    sb[idx] = v;
}

// ---------------------------------------------------------------------------
// Per class w: G = sb_w sb_w^T (5x5), Minv = (G + lam I)^-1 (Gauss-Jordan).
// 21 blocks x 128 threads.
// ---------------------------------------------------------------------------
__global__ __launch_bounds__(128)
void gram_inv_kernel(const float* __restrict__ sb, const float* __restrict__ r,
                     float* __restrict__ G, float* __restrict__ Minv)
{
    __shared__ float red[4][15];
    const int w   = blockIdx.x;
    const int tid = threadIdx.x;
    const float* S = sb + (size_t)w * SHOT * CDIM;

    float part[15];
    #pragma unroll
    for (int p = 0; p < 15; ++p) part[p] = 0.0f;

    for (int c = tid; c < CDIM; c += 128) {
        float v[5];
        #pragma unroll
        for (int s = 0; s < 5; ++s) v[s] = S[(size_t)s * CDIM + c];
        int p = 0;
        #pragma unroll
        for (int s = 0; s < 5; ++s)
            #pragma unroll
            for (int t = s; t < 5; ++t)
                part[p++] += v[s] * v[t];
    }

    #pragma unroll
    for (int off = 16; off > 0; off >>= 1)
        #pragma unroll
        for (int p = 0; p < 15; ++p)
            part[p] += __shfl_down(part[p], off, 32);

    const int lane = tid & 31, wv = tid >> 5;
    if (lane == 0)
        for (int p = 0; p < 15; ++p) red[wv][p] = part[p];
    __syncthreads();

    if (tid == 0) {
        float g[15];
        #pragma unroll
        for (int p = 0; p < 15; ++p)
            g[p] = red[0][p] + red[1][p] + red[2][p] + red[3][p];

        float a[5][5], inv[5][5];
        int p = 0;
        for (int s = 0; s < 5; ++s)
            for (int t = s; t < 5; ++t) { a[s][t] = g[p]; a[t][s] = g[p]; ++p; }
        for (int s = 0; s < 5; ++s)
            for (int t = 0; t < 5; ++t) {
                G[(size_t)w * 25 + s * 5 + t] = a[s][t];
                inv[s][t] = (s == t) ? 1.0f : 0.0f;
            }
        const float lam = ((float)SHOT / (float)CDIM) * __expf(r[0]) + 1e-6f;
        for (int s = 0; s < 5; ++s) a[s][s] += lam;
        // Gauss-Jordan (SPD, no pivoting needed)
        for (int k = 0; k < 5; ++k) {
            const float pinv = 1.0f / a[k][k];
            for (int j = 0; j < 5; ++j) { a[k][j] *= pinv; inv[k][j] *= pinv; }
            for (int i = 0; i < 5; ++i) {
                if (i == k) continue;
                const float f = a[i][k];
                for (int j = 0; j < 5; ++j) {
                    a[i][j]   -= f * a[k][j];
                    inv[i][j] -= f * inv[k][j];
                }
            }
        }
        for (int s = 0; s < 5; ++s)
            for (int t = 0; t < 5; ++t)
                Minv[(size_t)w * 25 + s * 5 + t] = inv[s][t];
    }
}

// ---------------------------------------------------------------------------
// qnorm[row] = ||q[row]||^2 ; one wave per row.
// ---------------------------------------------------------------------------
__global__ __launch_bounds__(256)
void qnorm_kernel(const float* __restrict__ q, float* __restrict__ qn)
{
    const int row  = blockIdx.x * 8 + (threadIdx.x >> 5);
    const int lane = threadIdx.x & 31;
    const float* p = q + (size_t)row * CDIM;
    float s = 0.0f;
    for (int c = lane; c < CDIM; c += 32) { const float v = p[c]; s += v * v; }
    #pragma unroll
    for (int off = 16; off > 0; off >>= 1) s += __shfl_down(s, off, 32);
    if (lane == 0) qn[row] = s;
}

// ---------------------------------------------------------------------------
// scores[r,w] = -exp(scale) * (rho^2 * bGb - 2*rho*(a.b) + ||q_r||^2) / C
// where a = A[r, w*5 : w*5+5], b = a @ Minv[w].
// ---------------------------------------------------------------------------
__global__ __launch_bounds__(256)
void scores_kernel(const float* __restrict__ A, const float* __restrict__ G,
                   const float* __restrict__ Minv, const float* __restrict__ qn,
                   const float* __restrict__ r, const float* __restrict__ scale,
                   float* __restrict__ out)
{
    const int idx = blockIdx.x * blockDim.x + threadIdx.x;
    if (idx >= ROI * NCLS) return;
    const int ri = idx / NCLS;
    const int w  = idx % NCLS;

    float a[5], b[5];
    #pragma unroll
    for (int t = 0; t < 5; ++t) a[t] = A[(size_t)ri * SBPAD + w * 5 + t];
    #pragma unroll
    for (int t = 0; t < 5; ++t) {
        float acc = 0.0f;
        #pragma unroll
        for (int s = 0; s < 5; ++s) acc += a[s] * Minv[(size_t)w * 25 + s * 5 + t];
        b[t] = acc;
    }
    float dot = 0.0f, qb2 = 0.0f;
    #pragma unroll
    for (int t = 0; t < 5; ++t) dot += a[t] * b[t];
    #pragma unroll
    for (int s = 0; s < 5; ++s) {
        float acc = 0.0f;
        #pragma unroll
        for (int t = 0; t < 5; ++t) acc += G[(size_t)w * 25 + s * 5 + t] * b[t];
        qb2 += b[s] * acc;
    }
    const float rho = __expf(r[1]);
    const float eu  = (rho * rho * qb2 - 2.0f * rho * dot + qn[ri]) * (1.0f / (float)CDIM);
    out[idx] = -eu * __expf(scale[0]);
}

// ---------------------------------------------------------------------------
extern "C" void kernel_launch(void* const* d_in, const int* in_sizes, int n_in,
                              void* d_out, int out_size, void* d_ws, size_t ws_size,
                              hipStream_t stream)
{
    const float* support_fc = (const float*)d_in[0];
    const float* bg_fc      = (const float*)d_in[1];
    // d_in[2] (query_fc) is unused by the reference.
    const float* box_fc     = (const float*)d_in[3];
    const float* W_ae1      = (const float*)d_in[4];
    const float* b_ae1      = (const float*)d_in[5];
    const float* W_ae2      = (const float*)d_in[6];
    const float* b_ae2      = (const float*)d_in[7];
    const float* W_bbox     = (const float*)d_in[8];
    const float* b_bbox     = (const float*)d_in[9];
    const float* r          = (const float*)d_in[10];
    const float* scalep     = (const float*)d_in[11];

    float* out_scores  = (float*)d_out;                    // (2048, 21)
    float* out_bbox    = out_scores + (size_t)ROI * NCLS;  // (2048, 84)
    float* out_support = out_bbox + (size_t)ROI * NBOX;    // (20, 5, 1024) == s flat

    float* ws    = (float*)d_ws;
    float* h_box = ws;                               // 2048*1024
    float* qbuf  = h_box + (size_t)ROI * CDIM;       // 2048*1024
    float* h_s   = qbuf + (size_t)ROI * CDIM;        // 100*1024
    float* h_bg  = h_s + (size_t)WAY * SHOT * CDIM;  // 100*1024
    float* bg_o  = h_bg + (size_t)WAY * SHOT * CDIM; // 100*1024
    float* sb    = bg_o + (size_t)WAY * SHOT * CDIM; // 112*1024
    float* Abuf  = sb + (size_t)SBPAD * CDIM;        // 2048*112
    float* Gbuf  = Abuf + (size_t)ROI * SBPAD;       // 21*25 (pad 528)
    float* Mibuf = Gbuf + 528;                       // 21*25 (pad 528)
    float* qn    = Mibuf + 528;                      // 2048

    const dim3 blk(256);
    auto ggrid = [](int M, int N) { return dim3((N + 127) / 128, (M + 127) / 128); };
    const int NS = WAY * SHOT; // 100

    // ae layer 1
    gemm_xwt_kernel<CDIM><<<ggrid(ROI, CDIM), blk, 0, stream>>>(box_fc,     W_ae1, b_ae1, h_box, ROI, CDIM, CDIM, CDIM, CDIM);
    gemm_xwt_kernel<CDIM><<<ggrid(NS,  CDIM), blk, 0, stream>>>(support_fc, W_ae1, b_ae1, h_s,   NS,  CDIM, CDIM, CDIM, CDIM);
    gemm_xwt_kernel<CDIM><<<ggrid(NS,  CDIM), blk, 0, stream>>>(bg_fc,      W_ae1, b_ae1, h_bg,  NS,  CDIM, CDIM, CDIM, CDIM);
    // ae layer 2
    gemm_xwt_kernel<CDIM><<<ggrid(ROI, CDIM), blk, 0, stream>>>(h_box, W_ae2, b_ae2, qbuf,        ROI, CDIM, CDIM, CDIM, CDIM);
    gemm_xwt_kernel<CDIM><<<ggrid(NS,  CDIM), blk, 0, stream>>>(h_s,   W_ae2, b_ae2, out_support, NS,  CDIM, CDIM, CDIM, CDIM);
    gemm_xwt_kernel<CDIM><<<ggrid(NS,  CDIM), blk, 0, stream>>>(h_bg,  W_ae2, b_ae2, bg_o,        NS,  CDIM, CDIM, CDIM, CDIM);
    // bbox head
    gemm_xwt_kernel<CDIM><<<ggrid(ROI, NBOX), blk, 0, stream>>>(box_fc, W_bbox, b_bbox, out_bbox, ROI, NBOX, CDIM, CDIM, NBOX);
    // sb = [bg_mean ; s]
    build_sb_kernel<<<(SBPAD * CDIM + 255) / 256, blk, 0, stream>>>(out_support, bg_o, sb);
    // per-class 5x5 Gram + ridge inverse
    gram_inv_kernel<<<NCLS, 128, 0, stream>>>(sb, r, Gbuf, Mibuf);
    // A = q @ sb^T  (2048 x 105)
    gemm_xwt_kernel<CDIM><<<ggrid(ROI, SBROWS), blk, 0, stream>>>(qbuf, sb, nullptr, Abuf, ROI, SBROWS, CDIM, CDIM, SBPAD);
    // row norms of q
    qnorm_kernel<<<ROI / 8, blk, 0, stream>>>(qbuf, qn);
    // fused Woodbury distance -> scores
    scores_kernel<<<(ROI * NCLS + 255) / 256, blk, 0, stream>>>(Abuf, Gbuf, Mibuf, qn, r, scalep, out_scores);
}